// YOLOLayer_16449724744284
// MI455X (gfx1250) — compile-verified
//
#include <hip/hip_runtime.h>
#include <hip/hip_bf16.h>
#include <stdint.h>

// ---------------------------------------------------------------------------
// YOLO head decode for MI455X (gfx1250).
// Memory-bound transform+transpose: uses CDNA5 async global->LDS DMA path
// (GLOBAL_LOAD_ASYNC_TO_LDS_B128 + s_wait_asynccnt) and an LDS-staged
// transpose so both global loads and global stores are 128B-coalesced.
// ---------------------------------------------------------------------------

#define G        52
#define GG       (G * G)          // 2704
#define NA       3
#define NCH      85               // nC + 5
#define TILE     G                // one grid row per block
#define ELEMS    (NCH * TILE)     // 4420 floats per tile
#define QUADS    (ELEMS / 4)      // 1105 float4 transfers
#define QPR      (TILE / 4)       // 13 quads per channel row
#define NTHREADS 256

typedef __attribute__((ext_vector_type(4))) float v4f;

__device__ __forceinline__ float fast_exp(float x) {
    // exp(x) = exp2(x * log2(e)) -> v_exp_f32
    return __builtin_amdgcn_exp2f(x * 1.4426950408889634f);
}
__device__ __forceinline__ float fast_sigmoid(float x) {
    return __builtin_amdgcn_rcpf(1.0f + fast_exp(-x));
}

__device__ __forceinline__ void async_load_b128_to_lds(unsigned lds_byte_off,
                                                       const void* gptr) {
    // gfx1250 VGLOBAL async op: per-lane gather into LDS, tracked by ASYNCcnt.
    asm volatile("global_load_async_to_lds_b128 %0, %1, off"
                 :
                 : "v"(lds_byte_off), "v"((unsigned long long)(size_t)gptr)
                 : "memory");
}

__global__ __launch_bounds__(NTHREADS) void yolo_decode_kernel(
    const float* __restrict__ x,       // (B, nA*85, g, g)
    const int* __restrict__ img_dim_p, // scalar
    float* __restrict__ out)           // (B, nA*g*g, 85)
{
    __shared__ __align__(16) float lds_raw[ELEMS]; // [c][s_local], linear
    __shared__ __align__(16) float lds_out[ELEMS]; // [s_local][c]

    const int t   = blockIdx.x; // grid row   0..51
    const int a   = blockIdx.y; // anchor     0..2
    const int b   = blockIdx.z; // batch
    const int tid = threadIdx.x;

    // Scalar load early so SMEM latency overlaps the async DMA below.
    const float stride = (float)(*img_dim_p) / (float)G; // 8.0 for 416/52

    // ---- Phase 0: async DMA global -> LDS (raw tile, coalesced b128) ----
    const float* xb =
        x + ((size_t)b * (NA * NCH) + (size_t)a * NCH) * GG + (size_t)t * TILE;
    for (int q = tid; q < QUADS; q += NTHREADS) {
        const int c  = q / QPR;        // channel 0..84
        const int sq = q - c * QPR;    // quad within row
        const float* gp = xb + (size_t)c * GG + (size_t)(sq * 4);
        // Low 32 bits of a generic pointer to LDS == LDS byte offset.
        const unsigned lds_off = (unsigned)(size_t)(const void*)(lds_raw + q * 4);
        async_load_b128_to_lds(lds_off, gp);
    }
    asm volatile("s_wait_asynccnt 0x0" ::: "memory");
    __syncthreads();

    // ---- Phase 1: per-channel transform + transpose into lds_out ----
    const float AW[NA] = {10.0f, 16.0f, 33.0f};
    const float AH[NA] = {13.0f, 30.0f, 23.0f};
    // (anchor/stride) * stride — mirror reference arithmetic order.
    const float aw = (AW[a] / stride) * stride;
    const float ah = (AH[a] / stride) * stride;
    const float frow = (float)t;

    for (int idx = tid; idx < ELEMS; idx += NTHREADS) {
        const int c  = idx / TILE;       // channel
        const int sl = idx - c * TILE;   // column within grid row
        const float v = lds_raw[idx];
        float r;
        if (c >= 4) {
            r = fast_sigmoid(v);                              // conf + classes
        } else if (c == 0) {
            r = (fast_sigmoid(v) + (float)sl) * stride;       // x center
        } else if (c == 1) {
            r = (fast_sigmoid(v) + frow) * stride;            // y center
        } else if (c == 2) {
            r = fast_exp(v) * aw;                             // width
        } else {
            r = fast_exp(v) * ah;                             // height
        }
        lds_out[sl * NCH + c] = r; // stride 85 (odd) -> bank-conflict-free
    }
    __syncthreads();

    // ---- Phase 2: contiguous non-temporal float4 stores ----
    v4f* __restrict__ dst = (v4f*)(out + (((size_t)b * NA + a) * GG +
                                          (size_t)t * TILE) * NCH);
    const v4f* __restrict__ src = (const v4f*)lds_out;
    for (int i = tid; i < QUADS; i += NTHREADS) {
        __builtin_nontemporal_store(src[i], dst + i);
    }
}

extern "C" void kernel_launch(void* const* d_in, const int* in_sizes, int n_in,
                              void* d_out, int out_size, void* d_ws, size_t ws_size,
                              hipStream_t stream) {
    const float* x       = (const float*)d_in[0];
    const int*   img_dim = (const int*)d_in[1];
    float*       out     = (float*)d_out;

    const int B = in_sizes[0] / (NA * NCH * GG); // 64 for the reference setup

    dim3 grid(G, NA, B);
    yolo_decode_kernel<<<grid, NTHREADS, 0, stream>>>(x, img_dim, out);
}